// ROIPooling2dPytorch_47991964566223
// MI455X (gfx1250) — compile-verified
//
#include <hip/hip_runtime.h>

// ROI max-pooling (adaptive 7x7) for x:(4,64,50,50) f32, rois:(M,5) i32.
// One 256-thread block per ROI. The strided 3-D crop (ws x hs x 64ch,
// strides 50 / 2500 elems) is DMA'd into LDS with ONE gfx1250
// tensor_load_to_lds (TDM, TENSORcnt), then the 64*49 outputs are computed
// from LDS and stored coalesced.

#define C_   64
#define H_   50
#define W_   50
#define PH_  7
#define PW_  7
#define NOUT (C_ * PH_ * PW_)

typedef unsigned int uint32x4 __attribute__((ext_vector_type(4)));
typedef unsigned int uint32x8 __attribute__((ext_vector_type(8)));

__global__ __launch_bounds__(256) void roi_pool_tdm_kernel(
    const float* __restrict__ x, const int* __restrict__ rois,
    float* __restrict__ out, int M)
{
    const int roi = blockIdx.x;
    if (roi >= M) return;
    const int tid = threadIdx.x;

    // ROI decode: float32 multiply by 1/16 (exact) then trunc, matching the
    // reference's astype(int32) on non-negative coords.
    const int b  = rois[roi * 5 + 0];
    const int x0 = (int)((float)rois[roi * 5 + 1] * 0.0625f);
    const int y0 = (int)((float)rois[roi * 5 + 2] * 0.0625f);
    const int x1 = (int)((float)rois[roi * 5 + 3] * 0.0625f);
    const int y1 = (int)((float)rois[roi * 5 + 4] * 0.0625f);
    const int hs = y1 - y0 + 1;   // >= 1
    const int ws = x1 - x0 + 1;   // >= 1
    const int chw = hs * ws;      // packed per-channel tile size in LDS

    __shared__ float tile[C_ * 64];   // 16 KB, holds 64 * hs * ws (<= 64*64)

    const bool useLds = (hs <= 8) && (ws <= 8);  // always true for this harness

    if (useLds && tid < 32) {
        // ---- Build Tensor DMA Descriptor (D#), ISA cdna5 ch.8 ----
        // Tile = (dim0=ws, dim1=hs, dim2=64 channels) of 4-byte elements,
        // global strides: dim0_stride=W (row), dim1_stride=H*W (channel).
        const unsigned long long gaddr = (unsigned long long)(const void*)
            (x + (((size_t)b * C_) * H_ + (size_t)y0) * W_ + x0);
        const unsigned ldsAddr = (unsigned)(unsigned long long)(const void*)&tile[0];

        uint32x4 g0;
        g0.x = 1u;                                  // count=1 (valid), flags 0
        g0.y = ldsAddr;                             // lds_addr [63:32]
        g0.z = (unsigned)(gaddr & 0xffffffffull);   // global_addr [95:64]
        g0.w = (unsigned)((gaddr >> 32) & 0x01ffffffull)  // global_addr [120:96]
             | (2u << 30);                          // type=2 ("image") [127:126]

        uint32x8 g1;
        g1.s0 = (2u << 16);                         // wg_mask=0, data_size=2 (4B)
        g1.s1 = ((unsigned)ws & 0xffffu) << 16;     // tensor_dim0[15:0] @ [63:48]
        g1.s2 = (((unsigned)ws >> 16) & 0xffffu)    // tensor_dim0[31:16]
              | (((unsigned)hs & 0xffffu) << 16);   // tensor_dim1[15:0]
        g1.s3 = (((unsigned)hs >> 16) & 0xffffu)    // tensor_dim1[31:16]
              | (((unsigned)ws & 0xffffu) << 16);   // tile_dim0 = ws
        g1.s4 = ((unsigned)hs & 0xffffu)            // tile_dim1 = hs
              | (64u << 16);                        // tile_dim2 = 64 channels
        g1.s5 = (unsigned)W_;                       // tensor_dim0_stride[31:0] = 50
        g1.s6 = 0u                                  // dim0_stride[47:32] = 0
              | (((unsigned)(H_ * W_) & 0xffffu) << 16); // dim1_stride[15:0]
        g1.s7 = ((unsigned)(H_ * W_) >> 16);        // dim1_stride[47:16] (=0)

        uint32x4 g2;
        g2.x = 64u;                                 // tensor_dim2 = 64
        g2.y = 0u; g2.z = 0u; g2.w = 0u;            // no dim3 / no iterate

        uint32x4 g3;
        g3.x = 0u; g3.y = 0u; g3.z = 0u; g3.w = 0u; // no dim3 stride / dim4

        // One DMA for the whole ROI crop (all 64 channels). TENSORcnt path.
        asm volatile("tensor_load_to_lds %0, %1, %2, %3"
                     :: "s"(g0), "s"(g1), "s"(g2), "s"(g3)
                     : "memory");
        __builtin_amdgcn_s_wait_tensorcnt(0);
    }
    __syncthreads();   // publish LDS tile to all 8 waves

    float* dst = out + (size_t)roi * NOUT;
    for (int o = tid; o < NOUT; o += 256) {
        const int c  = o / (PH_ * PW_);
        const int r  = o - c * (PH_ * PW_);
        const int ph = r / PW_;
        const int pw = r - ph * PW_;
        // Adaptive pooling bins: lo = p*size/P, hi = ceil((p+1)*size/P)
        const int ys = (ph * hs) / PH_;
        const int ye = ((ph + 1) * hs + PH_ - 1) / PH_;
        const int xs = (pw * ws) / PW_;
        const int xe = ((pw + 1) * ws + PW_ - 1) / PW_;

        float m = -3.402823466e+38f;  // finfo(f32).min
        if (useLds) {
            const float* t = &tile[c * chw];   // packed [c][y][x], row len ws
            for (int yy = ys; yy < ye; ++yy)
                for (int xx = xs; xx < xe; ++xx)
                    m = fmaxf(m, t[yy * ws + xx]);
        } else {
            // General fallback: read the crop straight from global.
            const float* src =
                x + (((size_t)b * C_ + c) * H_ + (size_t)y0) * W_ + x0;
            for (int yy = ys; yy < ye; ++yy)
                for (int xx = xs; xx < xe; ++xx)
                    m = fmaxf(m, src[yy * W_ + xx]);
        }
        dst[o] = m;   // coalesced: out[roi][c][ph][pw]
    }
}

extern "C" void kernel_launch(void* const* d_in, const int* in_sizes, int n_in,
                              void* d_out, int out_size, void* d_ws, size_t ws_size,
                              hipStream_t stream) {
    const float* x    = (const float*)d_in[0];
    const int*   rois = (const int*)d_in[1];
    float*       out  = (float*)d_out;
    const int M = in_sizes[1] / 5;   // rois is (M,5)
    if (M <= 0) return;
    roi_pool_tdm_kernel<<<dim3(M), dim3(256), 0, stream>>>(x, rois, out, M);
}